// AttentionA_89283780149276
// MI455X (gfx1250) — compile-verified
//
#include <hip/hip_runtime.h>
#include <math.h>

// Problem constants (match reference)
#define DM    1024        // model dim
#define NH    16          // heads
#define HD    64          // head dim
#define BB    2           // batch
#define NN    2048        // seq len
#define EPSV  1e-5f
#define GKC   32          // GEMM K-chunk staged in LDS
#define BPAD  160         // LDS pair-row stride (== 32 mod 64 -> halves on disjoint banks)

typedef float v2f __attribute__((ext_vector_type(2)));
typedef float v8f __attribute__((ext_vector_type(8)));

// D(16x16 f32) = A(16x4 f32) * B(4x16 f32) + C   -- full-precision fp32 matrix pipe
static __device__ __forceinline__ v8f wmma4(v2f a, v2f b, v8f c) {
  return __builtin_amdgcn_wmma_f32_16x16x4_f32(false, a, false, b, (short)0, c, false, false);
}

// CDNA5 async global->LDS b32 (ASYNCcnt); scalar fallback if builtins absent.
#if defined(__AMDGCN__) && __has_builtin(__builtin_amdgcn_global_load_async_to_lds_b32) && \
    __has_builtin(__builtin_amdgcn_s_wait_asynccnt)
typedef __attribute__((address_space(1))) int g_i32;
typedef __attribute__((address_space(3))) int l_i32;
static __device__ __forceinline__ void lds_fill4(const float* g, float* l) {
  __builtin_amdgcn_global_load_async_to_lds_b32((g_i32*)g, (l_i32*)l, 0, 0);
}
static __device__ __forceinline__ void lds_fill_wait() {
  __builtin_amdgcn_s_wait_asynccnt(0);
}
#else
static __device__ __forceinline__ void lds_fill4(const float* g, float* l) { *l = *g; }
static __device__ __forceinline__ void lds_fill_wait() {}
#endif

// ---------------------------------------------------------------- LayerNorm
__global__ void ln_kernel(const float* __restrict__ x, const float* __restrict__ w,
                          const float* __restrict__ bia, float* __restrict__ xn) {
  int row = blockIdx.x;
  const float* xr = x + (size_t)row * DM;
  float s = 0.f, s2 = 0.f;
  for (int i = threadIdx.x; i < DM; i += blockDim.x) { float v = xr[i]; s += v; s2 += v * v; }
  __shared__ float red0[8], red1[8];
  for (int off = 16; off; off >>= 1) { s += __shfl_down(s, off, 32); s2 += __shfl_down(s2, off, 32); }
  int wv = threadIdx.x >> 5, ln = threadIdx.x & 31;
  if (ln == 0) { red0[wv] = s; red1[wv] = s2; }
  __syncthreads();
  if (wv == 0) {
    s = (ln < 8) ? red0[ln] : 0.f; s2 = (ln < 8) ? red1[ln] : 0.f;
    for (int off = 4; off; off >>= 1) { s += __shfl_down(s, off, 32); s2 += __shfl_down(s2, off, 32); }
    if (ln == 0) { red0[0] = s; red1[0] = s2; }
  }
  __syncthreads();
  float mu = red0[0] * (1.0f / DM);
  float var = red1[0] * (1.0f / DM) - mu * mu;
  float rstd = rsqrtf(var + EPSV);
  float* xo = xn + (size_t)row * DM;
  for (int i = threadIdx.x; i < DM; i += blockDim.x)
    xo[i] = (xr[i] - mu) * rstd * w[i] + bia[i];
}

// ------------------------------------------------- fp32 WMMA GEMM  C = A @ B
// 256 threads = 8 waves; C macro-tile 128x64. B chunks (32x64) staged in LDS in
// a pair-interleaved layout: element (k,n) -> Bl[(k>>1)*BPAD + n*2 + (k&1)], so
// each B fragment is ONE aligned ds_load_b64 (no register marshaling). Chunks
// are double-buffered: async fill of chunk i+1 overlaps the 32 WMMAs of chunk i.
__global__ void gemm_wmma(const float* __restrict__ A, const float* __restrict__ B,
                          float* __restrict__ C, int K, int lda, int ldb, int ldc) {
  __shared__ __align__(16) float Bl[2][(GKC / 2) * BPAD];
  int t = threadIdx.x;
  int wv = t >> 5, lane = t & 31;
  int half = lane >> 4, lrow = lane & 15;
  int m0 = blockIdx.y * 128 + wv * 16;
  int n0 = blockIdx.x * 64;
  v8f a0 = {}, a1 = {}, a2 = {}, a3 = {};
  const float* Arow = A + (size_t)(m0 + lrow) * lda + half * 2;
  const float* Bg = B + n0;

  // cooperative async fill of one 32x64 chunk into pair-interleaved LDS
  auto fill = [&](int bufi, int k0) {
    const float* src = Bg + (size_t)k0 * ldb;
    float* dst = &Bl[bufi][0];
    #pragma unroll
    for (int e = 0; e < 8; e++) {
      int f = e * 256 + t;              // 0..2047
      int row = f >> 6, col = f & 63;   // 32 rows x 64 cols, 128B/wave coalesced
      lds_fill4(src + (size_t)row * ldb + col,
                dst + (row >> 1) * BPAD + col * 2 + (row & 1));
    }
  };

  fill(0, 0);
  int buf = 0;
  for (int k0 = 0; k0 < K; k0 += GKC) {
    lds_fill_wait();                    // my async fills for Bl[buf] done
    __syncthreads();                    // everyone's done (also WAR-protects refill)
    if (k0 + GKC < K) fill(buf ^ 1, k0 + GKC);   // overlap with compute below
    v2f af[8];
    #pragma unroll
    for (int i = 0; i < 8; i++) af[i] = *(const v2f*)(Arow + k0 + i * 4);
    const float* cur = &Bl[buf][0];
    #pragma unroll
    for (int kk = 0; kk < GKC; kk += 4) {
      int kp = (kk >> 1) + half;        // pair row of (kk + half*2)
      const float* bl = cur + kp * BPAD + lrow * 2;
      v2f b0 = *(const v2f*)(bl);       // (B[k][n], B[k+1][n]) contiguous
      v2f b1 = *(const v2f*)(bl + 32);
      v2f b2 = *(const v2f*)(bl + 64);
      v2f b3 = *(const v2f*)(bl + 96);
      v2f a = af[kk >> 2];
      a0 = wmma4(a, b0, a0); a1 = wmma4(a, b1, a1);
      a2 = wmma4(a, b2, a2); a3 = wmma4(a, b3, a3);
    }
    buf ^= 1;
  }
  float* Crow = C + (size_t)(m0 + half * 8) * ldc + n0 + lrow;
  #pragma unroll
  for (int r = 0; r < 8; r++) {
    Crow[(size_t)r * ldc + 0]  = a0[r];
    Crow[(size_t)r * ldc + 16] = a1[r];
    Crow[(size_t)r * ldc + 32] = a2[r];
    Crow[(size_t)r * ldc + 48] = a3[r];
  }
}

// ---------------------------------------------------------------- RoPE on q
__global__ void rope_kernel(float* __restrict__ q) {
  int idx = blockIdx.x * blockDim.x + threadIdx.x;      // B*N*H*32 = 2^21
  if (idx >= BB * NN * NH * (HD / 2)) return;
  int p = idx & 31;
  int h = (idx >> 5) & 15;
  int n = (idx >> 9) & (NN - 1);
  int b = idx >> 20;
  const float TOP = 2595.0f * log10f(21.0f);            // 2595*log10(1+4000/200)
  float lin = TOP * (float)p * (1.0f / 31.0f);          // linspace over 32 pts
  float mel = expf(lin * (2.3025850929940457f / 2595.0f)) - 1.0f;  // 10^(lin/2595)-1
  float freq = (36000.0f / 220.0f) * 0.2f * mel;
  float sv, cv;
  sincosf((float)n * freq, &sv, &cv);                   // precise range reduction
  size_t base = ((size_t)(b * NN + n)) * DM + h * HD + 2 * p;
  float q0 = q[base], q1 = q[base + 1];
  q[base]     = q0 * cv - q1 * sv;
  q[base + 1] = q1 * cv + q0 * sv;
}

// ------------------------- column stats:  m_j = max_i S[i,j], rz_j = 1/sum_i exp
__global__ void colstats_kernel(const float* __restrict__ q, const float* __restrict__ kv,
                                float* __restrict__ mcol, float* __restrict__ rzcol) {
  int wid = blockIdx.x * (blockDim.x >> 5) + (threadIdx.x >> 5);
  int lane = threadIdx.x & 31; int half = lane >> 4, lrow = lane & 15;
  int jt = wid & 127;           // N/16
  int bh = wid >> 7;            // b*NH+h
  int h = bh & 15, b = bh >> 4;
  const float* Qp = q  + (size_t)b * NN * DM + h * HD;
  const float* Kp = kv + (size_t)b * NN * (2 * DM) + h * HD;
  int j0 = jt * 16;
  float m[8], l[8];
  #pragma unroll
  for (int r = 0; r < 8; r++) { m[r] = -INFINITY; l[r] = 0.f; }
  const float* Krow = Kp + (size_t)(j0 + lrow) * (2 * DM) + half * 2;
  for (int i0 = 0; i0 < NN; i0 += 16) {
    v8f acc = {};
    const float* Qrow = Qp + (size_t)(i0 + lrow) * DM + half * 2;
    #pragma unroll
    for (int kk = 0; kk < HD; kk += 4) {
      v2f a  = *(const v2f*)(Krow + kk);     // K[j][k]  (rows of S^T)
      v2f bf = *(const v2f*)(Qrow + kk);     // Q[i][k]  (= (Q^T)[k][i])
      acc = wmma4(a, bf, acc);
    }
    #pragma unroll
    for (int r = 0; r < 8; r++) {            // online max/sum over i (per-lane stripe)
      float x = acc[r];
      float mn = fmaxf(m[r], x);
      l[r] = l[r] * __expf(m[r] - mn) + __expf(x - mn);
      m[r] = mn;
    }
  }
  #pragma unroll
  for (int r = 0; r < 8; r++) {              // combine 16 lanes of each half-wave
    float mm = m[r], ll = l[r];
    #pragma unroll
    for (int off = 8; off; off >>= 1) {
      float mo = __shfl_xor(mm, off, 32);
      float lo = __shfl_xor(ll, off, 32);
      float mn = fmaxf(mm, mo);
      ll = ll * __expf(mm - mn) + lo * __expf(mo - mn);
      mm = mn;
    }
    if (lrow == 0) {
      size_t o = (size_t)bh * NN + j0 + half * 8 + r;
      mcol[o]  = mm;
      rzcol[o] = 1.0f / ll;
    }
  }
}

// --------------- flash rows: dir-1 online-softmax attention + dir-2 column sums
__global__ void flash_kernel(const float* __restrict__ q, const float* __restrict__ kv,
                             const float* __restrict__ mcol, const float* __restrict__ rzcol,
                             float* __restrict__ xup, float* __restrict__ srow) {
  __shared__ float plds[4][16 * 18];         // per-wave P tile, pad-18 (conflict-free, 8B aligned)
  int wv = threadIdx.x >> 5;
  int wid = blockIdx.x * 4 + wv;
  int lane = threadIdx.x & 31; int half = lane >> 4, lrow = lane & 15;
  int it = wid & 127; int bh = wid >> 7; int h = bh & 15, b = bh >> 4;
  const float* Qp = q  + (size_t)b * NN * DM + h * HD;
  const float* Kp = kv + (size_t)b * NN * (2 * DM) + h * HD;
  const float* Vp = Kp + DM;
  const float* mc = mcol  + (size_t)bh * NN;
  const float* rz = rzcol + (size_t)bh * NN;
  int i0 = it * 16;
  float mi[8], li[8], sa[8];
  v8f o0 = {}, o1 = {}, o2 = {}, o3 = {};
  #pragma unroll
  for (int r = 0; r < 8; r++) { mi[r] = -INFINITY; li[r] = 0.f; sa[r] = 0.f; }
  const float* Qrow = Qp + (size_t)(i0 + lrow) * DM + half * 2;
  float* pw = &plds[wv][0];

  for (int j0 = 0; j0 < NN; j0 += 16) {
    // ---- S tile (16 i x 16 j): A = Q rows, B = K^T  (both aligned b64 loads)
    v8f acc = {};
    const float* Krow = Kp + (size_t)(j0 + lrow) * (2 * DM) + half * 2;
    #pragma unroll
    for (int kk = 0; kk < HD; kk += 4) {
      v2f a  = *(const v2f*)(Qrow + kk);
      v2f bf = *(const v2f*)(Krow + kk);
      acc = wmma4(a, bf, acc);
    }
    // ---- dir 2: s[i] += exp(S[i,j] - m_j) / Z_j  (stats precomputed)
    float mcj = mc[j0 + lrow], rzj = rz[j0 + lrow];
    #pragma unroll
    for (int r = 0; r < 8; r++) sa[r] += __expf(acc[r] - mcj) * rzj;
    // ---- dir 1: online row softmax; stage P through LDS for the A-fragment of P@V
    #pragma unroll
    for (int r = 0; r < 8; r++) {
      float x = acc[r];
      float tm = x;
      #pragma unroll
      for (int off = 8; off; off >>= 1) tm = fmaxf(tm, __shfl_xor(tm, off, 32));
      float mn = fmaxf(mi[r], tm);
      float p = __expf(x - mn);
      float ts = p;
      #pragma unroll
      for (int off = 8; off; off >>= 1) ts += __shfl_xor(ts, off, 32);
      float sc = __expf(mi[r] - mn);
      li[r] = li[r] * sc + ts;
      mi[r] = mn;
      o0[r] *= sc; o1[r] *= sc; o2[r] *= sc; o3[r] *= sc;
      pw[(half * 8 + r) * 18 + lrow] = p;    // C-layout -> LDS
    }
    // ---- O += P @ V   (A = P from LDS in A-layout, B = V rows)
    #pragma unroll
    for (int kk = 0; kk < 16; kk += 4) {
      int kb = kk + half * 2;
      v2f a = *(const v2f*)(pw + lrow * 18 + kb);
      const float* vr = Vp + (size_t)(j0 + kb) * (2 * DM) + lrow;
      v2f b0 = { vr[0],  vr[2 * DM] };
      v2f b1 = { vr[16], vr[2 * DM + 16] };
      v2f b2 = { vr[32], vr[2 * DM + 32] };
      v2f b3 = { vr[48], vr[2 * DM + 48] };
      o0 = wmma4(a, b0, o0); o1 = wmma4(a, b1, o1);
      o2 = wmma4(a, b2, o2); o3 = wmma4(a, b3, o3);
    }
  }
  // ---- epilogue: normalize dir-1 rows, reduce dir-2 row sums
  float* orow = xup + (size_t)(b * NN + i0 + half * 8) * DM + h * HD + lrow;
  #pragma unroll
  for (int r = 0; r < 8; r++) {
    float inv = 1.0f / li[r];
    orow[(size_t)r * DM + 0]  = o0[r] * inv;
    orow[(size_t)r * DM + 16] = o1[r] * inv;
    orow[(size_t)r * DM + 32] = o2[r] * inv;
    orow[(size_t)r * DM + 48] = o3[r] * inv;
    float ss = sa[r];
    #pragma unroll
    for (int off = 8; off; off >>= 1) ss += __shfl_xor(ss, off, 32);
    if (lrow == 0) srow[(size_t)bh * NN + i0 + half * 8 + r] = ss;
  }
}

// ------------------------------------------------ xa = s (per i,head) * v
__global__ void xa_scale_kernel(const float* __restrict__ kv, const float* __restrict__ srow,
                                float* __restrict__ xa) {
  int idx = blockIdx.x * blockDim.x + threadIdx.x;   // B*N*DM = 2^22
  if (idx >= BB * NN * DM) return;
  int d = idx & (DM - 1);
  int n = (idx >> 10) & (NN - 1);
  int b = idx >> 21;
  int h = d >> 6;
  float v = kv[((size_t)(b * NN + n)) * (2 * DM) + DM + d];
  float s = srow[((size_t)(b * NH + h)) * NN + n];
  xa[idx] = v * s;
}

// ---------------------------------------------------------------- launcher
extern "C" void kernel_launch(void* const* d_in, const int* in_sizes, int n_in,
                              void* d_out, int out_size, void* d_ws, size_t ws_size,
                              hipStream_t stream) {
  const float* x    = (const float*)d_in[0];
  const float* ln_w = (const float*)d_in[1];
  const float* ln_b = (const float*)d_in[2];
  const float* Wq   = (const float*)d_in[3];
  const float* Wkv  = (const float*)d_in[4];
  const float* Wout = (const float*)d_in[5];
  float* out = (float*)d_out;

  const size_t M = (size_t)BB * NN;          // 4096 rows
  float* ws   = (float*)d_ws;
  float* xn   = ws;                          // M*DM
  float* qb   = xn  + M * DM;                // M*DM
  float* kvb  = qb  + M * DM;                // M*2*DM
  float* xup  = kvb + M * 2 * DM;            // M*DM
  float* xab  = xup + M * DM;                // M*DM
  float* mco  = xab + M * DM;                // BB*NH*NN
  float* rzc  = mco + (size_t)BB * NH * NN;
  float* sro  = rzc + (size_t)BB * NH * NN;

  // 1) LayerNorm
  ln_kernel<<<M, 256, 0, stream>>>(x, ln_w, ln_b, xn);
  // 2) q = xn @ Wq ; kv = xn @ Wkv   (fp32 WMMA GEMMs, async double-buffered LDS B)
  gemm_wmma<<<dim3(DM / 64, M / 128), 256, 0, stream>>>(xn, Wq,  qb,  DM, DM, DM,     DM);
  gemm_wmma<<<dim3(2 * DM / 64, M / 128), 256, 0, stream>>>(xn, Wkv, kvb, DM, DM, 2 * DM, 2 * DM);
  // 3) RoPE on q
  rope_kernel<<<(BB * NN * NH * (HD / 2)) / 256, 256, 0, stream>>>(qb);
  // 4) column softmax stats (dir 2)
  colstats_kernel<<<(BB * NH * (NN / 16)) / 4, 128, 0, stream>>>(qb, kvb, mco, rzc);
  // 5) flash rows: dir-1 attention output + dir-2 column sums
  flash_kernel<<<(BB * NH * (NN / 16)) / 4, 128, 0, stream>>>(qb, kvb, mco, rzc, xup, sro);
  // 6) xa = s * v
  xa_scale_kernel<<<(BB * NN * DM) / 256, 256, 0, stream>>>(kvb, sro, xab);
  // 7) two output projections
  gemm_wmma<<<dim3(DM / 64, M / 128), 256, 0, stream>>>(xup, Wout, out,          DM, DM, DM, DM);
  gemm_wmma<<<dim3(DM / 64, M / 128), 256, 0, stream>>>(xab, Wout, out + M * DM, DM, DM, DM, DM);
}